// CRFFieldTypeClassification_66065186947130
// MI455X (gfx1250) — compile-verified
//
#include <hip/hip_runtime.h>
#include <hip/hip_bf16.h>

// Problem constants (from reference)
#define BATCH 64
#define SEQL  512
#define CH    1024
#define NTAGS 7          // 5 classes + START(5) + STOP(6)
#define NTOK  (BATCH*SEQL)   // 32768
#define NEGV  (-10000.0f)

typedef __attribute__((ext_vector_type(2))) float v2f;
typedef __attribute__((ext_vector_type(8))) float v8f;

// ---------------------------------------------------------------------------
// Kernel 1: pred = emb @ W^T + b  via V_WMMA_F32_16X16X4_F32, plus label copy.
// 256 threads = 8 waves; each wave computes one 16-row x 16-col (7 valid) tile.
// Grid of 256 blocks covers 256*8*16 = 32768 rows.
// ---------------------------------------------------------------------------
__global__ __launch_bounds__(256) void crf_gemm_kernel(
    const float* __restrict__ A,      // [32768, 1024] embeddings
    const float* __restrict__ W,      // [7, 1024]
    const float* __restrict__ bias,   // [7]
    const int*   __restrict__ lab,    // [32768]
    float*       __restrict__ out,    // [32768, 7] pred region of d_out
    int*         __restrict__ labOut) // [32768] int32 region of d_out
{
    const int lane  = threadIdx.x & 31;
    const int wave  = threadIdx.x >> 5;
    const int rowBase = (blockIdx.x * 8 + wave) * 16;

    const int mcol  = lane & 15;          // A: row-in-tile; B/C: col (tag)
    const int khalf = (lane >> 4) * 2;    // 0 for lanes 0-15, 2 for lanes 16-31

    const float* Arow = A + (size_t)(rowBase + mcol) * CH + khalf;
    const int    n    = mcol;
    const float  wmask = (n < NTAGS) ? 1.0f : 0.0f;
    const float* Wrow  = W + (size_t)((n < NTAGS) ? n : 0) * CH + khalf;

    v8f acc = {};   // C accumulator, zero-init

    #pragma unroll 4
    for (int k = 0; k < CH; k += 4) {
        v2f av; av.x = Arow[k];          av.y = Arow[k + 1];
        v2f bv; bv.x = Wrow[k] * wmask;  bv.y = Wrow[k + 1] * wmask;
        // 8 args: (neg_a, A, neg_b, B, c_mod, C, reuse_a, reuse_b)
        acc = __builtin_amdgcn_wmma_f32_16x16x4_f32(
            false, av, false, bv, (short)0, acc, false, false);
    }

    // C/D layout: vgpr r, lanes 0-15 -> M=r, lanes 16-31 -> M=r+8; N = lane%16
    if (n < NTAGS) {
        const float bs = bias[n];
        const int rh = (lane >> 4) * 8;
        #pragma unroll
        for (int r = 0; r < 8; ++r) {
            out[(size_t)(rowBase + rh + r) * NTAGS + n] = acc[r] + bs;
        }
    }

    // label_class passthrough as int32 (128 rows per block)
    if (threadIdx.x < 128) {
        const int idx = blockIdx.x * 128 + threadIdx.x;
        labOut[idx] = lab[idx];
    }
}

// ---------------------------------------------------------------------------
// Kernel 2: CRF negative log-likelihood, summed over batch.
// One workgroup, 512 threads = 64 sequences x 8 lanes.
// Lanes 0..6 of each group carry forward-var for tag==lane (tag-parallel
// log-sum-exp recurrence, fv exchanged with ds_bpermute shuffles).
// Lane 7 computes the gold path score. Final batch reduction in LDS.
// ---------------------------------------------------------------------------
__global__ __launch_bounds__(512) void crf_nll_kernel(
    const float* __restrict__ feats,  // [64, 512, 7] (pred region of d_out)
    const int*   __restrict__ labels, // [64, 512]
    const float* __restrict__ trans,  // [7, 7]  trans[next*7 + prev]
    float*       __restrict__ score)  // d_out[0]
{
    __shared__ float tr[NTAGS * NTAGS];
    __shared__ float partial[BATCH];

    const int t = threadIdx.x;
    if (t < NTAGS * NTAGS) tr[t] = trans[t];
    __syncthreads();

    const int seq = t >> 3;
    const int tag = t & 7;                 // 0..6 real tags, 7 = gold lane
    const int myTag = (tag < NTAGS) ? tag : 0;

    float trow[NTAGS];
    #pragma unroll
    for (int p = 0; p < NTAGS; ++p) trow[p] = tr[myTag * NTAGS + p];

    float fv = (tag == 5) ? 0.0f : NEGV;   // START = 5
    float gold = 0.0f;
    int prevTag = 5;

    const float* frow = feats + (size_t)seq * SEQL * NTAGS;
    const int*   lrow = labels + (size_t)seq * SEQL;

    for (int l = 0; l < SEQL; ++l) {
        // gather fv[prev] from the 8-lane group and add transition row
        float x[NTAGS];
        #pragma unroll
        for (int p = 0; p < NTAGS; ++p)
            x[p] = __shfl(fv, p, 8) + trow[p];

        float m = x[0];
        #pragma unroll
        for (int p = 1; p < NTAGS; ++p) m = fmaxf(m, x[p]);
        float s = 0.0f;
        #pragma unroll
        for (int p = 0; p < NTAGS; ++p) s += expf(x[p] - m);
        const float lse = m + logf(s);

        const float emit = frow[l * NTAGS + myTag];
        const int   tg   = lrow[l];

        if (tag == 7) {
            gold += frow[l * NTAGS + tg] + tr[tg * NTAGS + prevTag];
            prevTag = tg;
            fv = NEGV;                      // padding lane stays -inf-ish
        } else {
            fv = lse + emit;
        }
    }

    // terminal: fwd = lse_tag( fv[tag] + trans[STOP, tag] ), STOP = 6
    float xf = (tag < NTAGS) ? (fv + tr[6 * NTAGS + tag]) : NEGV;
    float m = xf;
    #pragma unroll
    for (int off = 1; off < 8; off <<= 1) m = fmaxf(m, __shfl_xor(m, off, 8));
    float s = expf(xf - m);
    #pragma unroll
    for (int off = 1; off < 8; off <<= 1) s += __shfl_xor(s, off, 8);
    const float fwd = m + logf(s);

    if (tag == 7) {
        gold += tr[6 * NTAGS + prevTag];   // trans[STOP, last_tag]
        partial[seq] = fwd - gold;
    }
    __syncthreads();

    if (t == 0) {
        float acc = 0.0f;
        #pragma unroll 8
        for (int i = 0; i < BATCH; ++i) acc += partial[i];
        score[0] = acc;
    }
}

// ---------------------------------------------------------------------------
extern "C" void kernel_launch(void* const* d_in, const int* in_sizes, int n_in,
                              void* d_out, int out_size, void* d_ws, size_t ws_size,
                              hipStream_t stream) {
    const float* emb   = (const float*)d_in[0];  // [32768,1024] f32
    const int*   lab   = (const int*)  d_in[1];  // [32768] i32
    const float* W     = (const float*)d_in[2];  // [7,1024] f32
    const float* bias  = (const float*)d_in[3];  // [7] f32
    const float* trans = (const float*)d_in[4];  // [7,7] f32

    float* outF   = (float*)d_out;
    int*   labOut = ((int*)d_out) + 1;            // output #2: int32 labels
    float* pred   = outF + 1 + NTOK;              // output #3: [32768,7] logits

    crf_gemm_kernel<<<256, 256, 0, stream>>>(emb, W, bias, lab, pred, labOut);
    crf_nll_kernel<<<1, 512, 0, stream>>>(pred, lab, trans, outF);
}